// SModel_11227044512394
// MI455X (gfx1250) — compile-verified
//
#include <hip/hip_runtime.h>

typedef float v2f __attribute__((ext_vector_type(2)));
typedef float v8f __attribute__((ext_vector_type(8)));

#define WAVES_PER_BLOCK 8
#define EDGE_BLOCKS 1024

__device__ __forceinline__ float leaky_f(float x) { return x > 0.f ? x : 0.1f * x; }

// Load a 15x15 row-major weight matrix, zero-padded to 16x16, as the four
// B operands (K-slices of 4) for V_WMMA_F32_16X16X4_F32.
// B layout (4x16, K x N): lanes 0-15 = N, VGPR0 holds K=kb+0 (lanes 0-15) /
// K=kb+2 (lanes 16-31), VGPR1 holds K=kb+1 / K=kb+3 (mirrors the documented
// 16x4 A layout).
__device__ __forceinline__ void load_B(const float* __restrict__ W, int lane, v2f b[4]) {
  int col = lane & 15;
  int half = lane >> 4;
#pragma unroll
  for (int q = 0; q < 4; ++q) {
    int k0 = 4 * q + 2 * half;
    int k1 = k0 + 1;
    float x = (k0 < 15 && col < 15) ? W[k0 * 15 + col] : 0.f;
    float y = (k1 < 15 && col < 15) ? W[k1 * 15 + col] : 0.f;
    v2f t = {x, y};
    b[q] = t;
  }
}

// Compute one 16-edge tile of the edge MLP:
//   out = leaky([x_t[tgt] | edge_attr] @ W1 + b1) @ W2 + b2
// Result in C/D layout: lane L, element r -> edge row m = r + 8*(L>>4),
// output column n = L & 15 (column 15 is padding and computes to 0).
__device__ __forceinline__ v8f edge_mlp_tile(
    long e0, long E,
    const float* __restrict__ x_t, const float* __restrict__ edge_attr,
    const int* __restrict__ tgt_idx,
    const v2f bw1[4], const v2f bw2[4], float bias1,
    float* __restrict__ ldsTile, int lane) {
  int row = lane & 15;
  int half = lane >> 4;
  long e = e0 + row;
  if (e >= E) e = E - 1;  // clamp: keep EXEC all-ones for WMMA
  int tg = tgt_idx[e];
  const float* xt = x_t + (long)tg * 5;
  const float* ea = edge_attr + e * 10;
  float f0 = xt[0], f1 = xt[1], f2 = xt[2], f3 = xt[3], f4 = xt[4];
  float f5 = ea[0], f6 = ea[1], f7 = ea[2], f8 = ea[3], f9 = ea[4];
  float f10 = ea[5], f11 = ea[6], f12 = ea[7], f13 = ea[8], f14 = ea[9];
  float f15 = 0.f;

  // A operands: 16x4 f32 tiles (lanes 0-15 rows; halves split K pairs)
  v2f a0 = {half ? f2 : f0, half ? f3 : f1};
  v2f a1 = {half ? f6 : f4, half ? f7 : f5};
  v2f a2 = {half ? f10 : f8, half ? f11 : f9};
  v2f a3 = {half ? f14 : f12, half ? f15 : f13};

  v8f acc = {};
  acc = __builtin_amdgcn_wmma_f32_16x16x4_f32(false, a0, false, bw1[0], (short)0, acc, false, false);
  acc = __builtin_amdgcn_wmma_f32_16x16x4_f32(false, a1, false, bw1[1], (short)0, acc, false, false);
  acc = __builtin_amdgcn_wmma_f32_16x16x4_f32(false, a2, false, bw1[2], (short)0, acc, false, false);
  acc = __builtin_amdgcn_wmma_f32_16x16x4_f32(false, a3, false, bw1[3], (short)0, acc, false, false);

  // bias + leaky; stage hidden tile in LDS (row stride 17 -> conflict-free)
#pragma unroll
  for (int r = 0; r < 8; ++r) {
    int m = r + 8 * half;
    ldsTile[m * 17 + (lane & 15)] = leaky_f(acc[r] + bias1);
  }
  asm volatile("s_wait_dscnt 0x0" ::: "memory");

  // re-read hidden tile in A layout
  int c = 2 * half;
  const float* lr = ldsTile + row * 17 + c;
  v2f h0 = {lr[0], lr[1]};
  v2f h1 = {lr[4], lr[5]};
  v2f h2 = {lr[8], lr[9]};
  v2f h3 = {lr[12], lr[13]};

  v8f acc2 = {};
  acc2 = __builtin_amdgcn_wmma_f32_16x16x4_f32(false, h0, false, bw2[0], (short)0, acc2, false, false);
  acc2 = __builtin_amdgcn_wmma_f32_16x16x4_f32(false, h1, false, bw2[1], (short)0, acc2, false, false);
  acc2 = __builtin_amdgcn_wmma_f32_16x16x4_f32(false, h2, false, bw2[2], (short)0, acc2, false, false);
  acc2 = __builtin_amdgcn_wmma_f32_16x16x4_f32(false, h3, false, bw2[3], (short)0, acc2, false, false);
  return acc2;
}

__global__ void zero_kernel(float* __restrict__ p, long n) {
  long i = (long)blockIdx.x * blockDim.x + threadIdx.x;
  long stride = (long)gridDim.x * blockDim.x;
  for (; i < n; i += stride) p[i] = 0.f;
}

__global__ void edge_pass1(const float* __restrict__ x_t, const float* __restrict__ edge_attr,
                           const int* __restrict__ tgt_idx, const int* __restrict__ src_idx,
                           const float* __restrict__ W1, const float* __restrict__ b1,
                           const float* __restrict__ W2, const float* __restrict__ b2,
                           float* __restrict__ cnt, float* __restrict__ sum,
                           float* __restrict__ sumsq, long E, long numTiles) {
  __shared__ float lds[WAVES_PER_BLOCK][16 * 17];
  __shared__ int lsrc[WAVES_PER_BLOCK][16];
  int lane = threadIdx.x & 31;
  int w = threadIdx.x >> 5;
  long wave = (long)blockIdx.x * WAVES_PER_BLOCK + w;
  long nwaves = (long)gridDim.x * WAVES_PER_BLOCK;
  int n = lane & 15, half = lane >> 4;

  v2f bw1[4], bw2[4];
  load_B(W1, lane, bw1);
  load_B(W2, lane, bw2);
  float bias1 = (n < 15) ? b1[n] : 0.f;
  float bias2 = (n < 15) ? b2[n] : 0.f;

  for (long t = wave; t < numTiles; t += nwaves) {
    long e0 = t * 16;
    {
      long e = e0 + n;
      if (e >= E) e = E - 1;
      lsrc[w][n] = src_idx[e];  // both halves write identical values
    }
    v8f out = edge_mlp_tile(e0, E, x_t, edge_attr, tgt_idx, bw1, bw2, bias1, &lds[w][0], lane);
    asm volatile("s_wait_dscnt 0x0" ::: "memory");
#pragma unroll
    for (int r = 0; r < 8; ++r) {
      int m = r + 8 * half;
      if (e0 + m < E) {
        int s = lsrc[w][m];
        float v = out[r] + bias2;
        atomicAdd(&sum[(long)s * 16 + n], v);
        atomicAdd(&sumsq[(long)s * 16 + n], v * v);
        if (n == 0) atomicAdd(&cnt[s], 1.f);
      }
    }
  }
}

__global__ void edge_pass2(const float* __restrict__ x_t, const float* __restrict__ edge_attr,
                           const int* __restrict__ tgt_idx, const int* __restrict__ src_idx,
                           const float* __restrict__ W1, const float* __restrict__ b1,
                           const float* __restrict__ W2, const float* __restrict__ b2,
                           const float* __restrict__ a_arr,
                           float* __restrict__ csum, float* __restrict__ dsum,
                           long E, long numTiles) {
  __shared__ float lds[WAVES_PER_BLOCK][16 * 17];
  __shared__ int lsrc[WAVES_PER_BLOCK][16];
  int lane = threadIdx.x & 31;
  int w = threadIdx.x >> 5;
  long wave = (long)blockIdx.x * WAVES_PER_BLOCK + w;
  long nwaves = (long)gridDim.x * WAVES_PER_BLOCK;
  int n = lane & 15, half = lane >> 4;

  v2f bw1[4], bw2[4];
  load_B(W1, lane, bw1);
  load_B(W2, lane, bw2);
  float bias1 = (n < 15) ? b1[n] : 0.f;
  float bias2 = (n < 15) ? b2[n] : 0.f;

  for (long t = wave; t < numTiles; t += nwaves) {
    long e0 = t * 16;
    {
      long e = e0 + n;
      if (e >= E) e = E - 1;
      lsrc[w][n] = src_idx[e];
    }
    v8f out = edge_mlp_tile(e0, E, x_t, edge_attr, tgt_idx, bw1, bw2, bias1, &lds[w][0], lane);
    asm volatile("s_wait_dscnt 0x0" ::: "memory");
#pragma unroll
    for (int r = 0; r < 8; ++r) {
      int m = r + 8 * half;
      if (e0 + m < E) {
        int s = lsrc[w][m];
        float diff = (out[r] + bias2) - a_arr[(long)s * 16 + n];
        float d2 = diff * diff;
        atomicAdd(&csum[(long)s * 16 + n], d2 * diff);
        atomicAdd(&dsum[(long)s * 16 + n], d2 * d2);
      }
    }
  }
}

__global__ void node_stats(const float* __restrict__ cnt, const float* __restrict__ sum,
                           const float* __restrict__ sumsq,
                           float* __restrict__ a_arr, float* __restrict__ b_arr, long NS) {
  long i = (long)blockIdx.x * blockDim.x + threadIdx.x;
  if (i >= NS) return;
  float denom = fmaxf(cnt[i], 1.f);
  float inv = 1.f / denom;
#pragma unroll
  for (int j = 0; j < 16; ++j) {
    float a = sum[i * 16 + j] * inv;
    float m2 = sumsq[i * 16 + j] * inv;
    float var = m2 - a * a;
    var = var > 0.f ? var : 0.f;
    a_arr[i * 16 + j] = a;
    b_arr[i * 16 + j] = sqrtf(1e-6f + var);
  }
}

__global__ void node_final(const float* __restrict__ x_s, const float* __restrict__ u,
                           const int* __restrict__ batch_s,
                           const float* __restrict__ cnt, const float* __restrict__ a_arr,
                           const float* __restrict__ b_arr, const float* __restrict__ csum,
                           const float* __restrict__ dsum,
                           const float* __restrict__ W3, const float* __restrict__ b3,
                           const float* __restrict__ W4, const float* __restrict__ b4,
                           float* __restrict__ out, long NS) {
  long i = (long)blockIdx.x * blockDim.x + threadIdx.x;
  if (i >= NS) return;
  float feat[81];
  const float* xs = x_s + i * 10;
#pragma unroll
  for (int j = 0; j < 10; ++j) feat[j] = xs[j];
  float nn = cnt[i];
  feat[10] = nn;
  float inv = 1.f / fmaxf(nn, 1.f);
#pragma unroll
  for (int j = 0; j < 15; ++j) {
    float a = a_arr[i * 16 + j];
    float b = b_arr[i * 16 + j];
    float b2 = b * b;
    feat[11 + j] = a;
    feat[26 + j] = b;
    feat[41 + j] = csum[i * 16 + j] * inv / (b2 * b);
    feat[56 + j] = dsum[i * 16 + j] * inv / (b2 * b2);
  }
  const float* uu = u + (long)batch_s[i] * 10;
#pragma unroll
  for (int j = 0; j < 10; ++j) feat[71 + j] = uu[j];

  float h[10];
#pragma unroll
  for (int j = 0; j < 10; ++j) {
    float acc = b3[j];
#pragma unroll
    for (int k = 0; k < 81; ++k) acc += feat[k] * W3[k * 10 + j];
    h[j] = leaky_f(acc);
  }
#pragma unroll
  for (int j = 0; j < 10; ++j) {
    float acc = b4[j];
#pragma unroll
    for (int k = 0; k < 10; ++k) acc += h[k] * W4[k * 10 + j];
    out[i * 10 + j] = acc;
  }
}

extern "C" void kernel_launch(void* const* d_in, const int* in_sizes, int n_in,
                              void* d_out, int out_size, void* d_ws, size_t ws_size,
                              hipStream_t stream) {
  const float* x_s = (const float*)d_in[0];
  const float* x_t = (const float*)d_in[1];
  const float* edge_attr = (const float*)d_in[2];
  const float* u = (const float*)d_in[3];
  const float* W1 = (const float*)d_in[4];
  const float* b1 = (const float*)d_in[5];
  const float* W2 = (const float*)d_in[6];
  const float* b2 = (const float*)d_in[7];
  const float* W3 = (const float*)d_in[8];
  const float* b3 = (const float*)d_in[9];
  const float* W4 = (const float*)d_in[10];
  const float* b4 = (const float*)d_in[11];
  const int* src_idx = (const int*)d_in[12];
  const int* tgt_idx = (const int*)d_in[13];
  const int* batch_s = (const int*)d_in[14];

  long NS = in_sizes[0] / 10;
  long E = in_sizes[2] / 10;
  float* out = (float*)d_out;

  // workspace layout (floats): cnt[NS] | sum[NS*16] | sumsq[NS*16] |
  // a[NS*16] | b[NS*16] | csum[NS*16] | dsum[NS*16]
  float* ws = (float*)d_ws;
  float* cnt = ws;
  float* sum = cnt + NS;
  float* sumsq = sum + NS * 16;
  float* a_arr = sumsq + NS * 16;
  float* b_arr = a_arr + NS * 16;
  float* csum = b_arr + NS * 16;
  float* dsum = csum + NS * 16;

  zero_kernel<<<1024, 256, 0, stream>>>(ws, NS * 97);

  long numTiles = (E + 15) / 16;
  dim3 blk(32 * WAVES_PER_BLOCK);

  edge_pass1<<<EDGE_BLOCKS, blk, 0, stream>>>(x_t, edge_attr, tgt_idx, src_idx,
                                              W1, b1, W2, b2, cnt, sum, sumsq, E, numTiles);
  node_stats<<<(int)((NS + 255) / 256), 256, 0, stream>>>(cnt, sum, sumsq, a_arr, b_arr, NS);
  edge_pass2<<<EDGE_BLOCKS, blk, 0, stream>>>(x_t, edge_attr, tgt_idx, src_idx,
                                              W1, b1, W2, b2, a_arr, csum, dsum, E, numTiles);
  node_final<<<(int)((NS + 255) / 256), 256, 0, stream>>>(x_s, u, batch_s, cnt, a_arr, b_arr,
                                                          csum, dsum, W3, b3, W4, b4, out, NS);
}